// AbstractSom_42245298324026
// MI455X (gfx1250) — compile-verified
//
#include <hip/hip_runtime.h>
#include <math.h>

#define XDIM 512
#define YDIM 512
#define DDIM 256
#define NCELLS (XDIM * YDIM)
#define SIGMA_F 12.0f
#define ALPHA_F 0.05f

// TDM tile geometry for the distance kernel
#define CHUNK 32                    // features per TDM chunk
#define ROWSTRIDE 36                // 32 data + 4 pad DWORDs (bank-conflict-free)
#define ROWS 128                    // cells (rows) per block
#define BUF_DW (ROWS * ROWSTRIDE)   // 4608 DWORDs = 18432 B per buffer

typedef float v2f __attribute__((ext_vector_type(2)));
typedef float v8f __attribute__((ext_vector_type(8)));
typedef unsigned int u32x4 __attribute__((ext_vector_type(4)));
typedef int i32x4 __attribute__((ext_vector_type(4)));
typedef int i32x8 __attribute__((ext_vector_type(8)));

struct Best2 { float v1; int i1; float v2; int i2; };

__device__ inline void tdm_issue_load(unsigned lds_addr, unsigned long long gaddr,
                                      const i32x8& g1) {
  u32x4 g0;
  g0[0] = 1u;                                              // count=1, user mode
  g0[1] = lds_addr;                                        // LDS byte address
  g0[2] = (unsigned)(gaddr & 0xffffffffu);                 // global_addr[31:0]
  g0[3] = (unsigned)((gaddr >> 32) & 0x01ffffffu)          // global_addr[56:32]
          | 0x80000000u;                                   // type=2 ("image")
  i32x4 gz4 = {0, 0, 0, 0};
  i32x8 gz8 = {0, 0, 0, 0, 0, 0, 0, 0};
  // 6-arg form (clang-23 / therock-10.0 headers): (g0, g1, g2, g3, g4, cpol)
  __builtin_amdgcn_tensor_load_to_lds(g0, g1, gz4, gz4, gz8, 0);
}

// ---------------------------------------------------------------------------
// Phase 1: per-cell p = ||w||^2 - 2*(w.v)   (constant ||v||^2 added later)
// TDM DMAs a 128-cell x 32-feature tile into LDS (double-buffered, padded
// rows -> conflict-free ds_load_b64), waves feed V_WMMA_F32_16X16X4_F32 with
// B columns all equal to the v-chunk so every D column = w.v in full f32.
// A layout (ISA 7.12.2, 32-bit A 16x4): lane m / m+16 hold cell m,
// VGPR0/1 = K {0,1} (lanes 0-15) or K {2,3} (lanes 16-31).
// ---------------------------------------------------------------------------
__global__ __launch_bounds__(256) void som_dist_kernel(
    const float* __restrict__ w, const float* __restrict__ v,
    float* __restrict__ p_out) {
  __shared__ float smem[2 * BUF_DW];          // 36 KB double buffer
  const int lane = threadIdx.x & 31;
  const int wave = threadIdx.x >> 5;
  const int cellblk = blockIdx.x * ROWS;      // first cell handled by block
  const int m = lane & 15;
  const int khalf = (lane >> 4) * 2;          // 0 or 2
  const int row = wave * 16 + m;              // row in the staged tile

  const unsigned lds_base = (unsigned)(uintptr_t)(void*)smem;
  const unsigned long long gbase =
      (unsigned long long)(uintptr_t)(const void*)w +
      (unsigned long long)cellblk * (DDIM * 4ull);

  // D# group 1 (loop-invariant): 4-byte elements, LDS pad 4 DW per 32 DW,
  // tensor 256 x 262144 (stride 256), tile 32 x 128.
  i32x8 g1;
  g1[0] = (2 << 16)                 // data_size = 4 bytes
        | (1 << 20)                 // pad_enable
        | (4 << 22)                 // pad_interval code 4 = 32 DWORDs
        | (3 << 25);                // pad_amount  code 3 = 4 DWORDs
  g1[1] = (DDIM & 0xffff) << 16;                                  // dim0 lo16
  g1[2] = ((DDIM >> 16) & 0xffff) | ((NCELLS & 0xffff) << 16);    // dim0 hi, dim1 lo
  g1[3] = ((NCELLS >> 16) & 0xffff) | (CHUNK << 16);              // dim1 hi, tile_dim0
  g1[4] = ROWS;                     // tile_dim1 = 128, tile_dim2 = 0
  g1[5] = DDIM;                     // tensor_dim0_stride lo32 = 256
  g1[6] = 0;                        // stride hi16, dim1_stride lo16
  g1[7] = 0;                        // dim1_stride hi32

  if (wave == 0) {                  // prime the pipeline with chunk 0
    tdm_issue_load(lds_base, gbase, g1);
  }

  v8f c = {};
  float ssq = 0.0f;
  for (int kc = 0; kc < DDIM / CHUNK; ++kc) {
    if (wave == 0) {
      __builtin_amdgcn_s_wait_tensorcnt(0);   // chunk kc resident in LDS
    }
    __syncthreads();   // (a) chunk kc visible to all, (b) chunk kc-1 consumed
    if (wave == 0 && kc + 1 < DDIM / CHUNK) { // prefetch chunk kc+1
      const unsigned long long ga =
          gbase + (unsigned long long)((kc + 1) * CHUNK) * 4ull;
      tdm_issue_load(lds_base + (unsigned)(((kc + 1) & 1) * (BUF_DW * 4)), ga, g1);
    }
    const float* buf = smem + (kc & 1) * BUF_DW + row * ROWSTRIDE + khalf;
    const float* vb = v + kc * CHUNK + khalf;
#pragma unroll
    for (int kb = 0; kb < CHUNK; kb += 4) {
      v2f a = *(const v2f*)(buf + kb);        // ds_load_b64, conflict-free
      v2f b = *(const v2f*)(vb + kb);
      ssq += a.x * a.x + a.y * a.y;
      // (neg_a, A, neg_b, B, c_mod, C, reuse_a, reuse_b)
      c = __builtin_amdgcn_wmma_f32_16x16x4_f32(false, a, false, b,
                                                (short)0, c, false, false);
    }
  }

  // full ||w||^2 for cell m lives split across lanes m and m+16
  ssq += __shfl_xor(ssq, 16);

  // D layout: lane n (n<16) holds rows m=0..7 in c[0..7]; lane 16+n rows 8..15.
  __shared__ float s_dot[8][16];
  __shared__ float s_ssq[8][16];
  if (lane < 16) s_ssq[wave][m] = ssq;
  if (m == 0) {
    const int base = (lane >> 4) * 8;
#pragma unroll
    for (int r = 0; r < 8; ++r) s_dot[wave][base + r] = c[r];
  }
  __syncthreads();
  if (lane < 16) {
    p_out[cellblk + wave * 16 + m] = s_ssq[wave][m] - 2.0f * s_dot[wave][m];
  }
}

// ---------------------------------------------------------------------------
// Top-2 (value, index) lexicographic reduction == jnp.argmin first-occurrence
// tie-breaking for winner AND for the "replace winner with max, argmin" 2nd.
// ---------------------------------------------------------------------------
__device__ inline void merge_one(float v, int i,
                                 float& v1, int& i1, float& v2, int& i2) {
  if (v < v1 || (v == v1 && i < i1)) {
    v2 = v1; i2 = i1; v1 = v; i1 = i;
  } else if (v < v2 || (v == v2 && i < i2)) {
    v2 = v; i2 = i;
  }
}

__global__ __launch_bounds__(256) void som_reduce1(
    const float* __restrict__ p, Best2* __restrict__ bres) {
  __shared__ float sv1[256]; __shared__ int si1[256];
  __shared__ float sv2[256]; __shared__ int si2[256];
  const int tid = threadIdx.x;
  float v1 = 3.4e38f, v2 = 3.4e38f;
  int i1 = 0x7fffffff, i2 = 0x7fffffff;
#pragma unroll
  for (int k = 0; k < 4; ++k) {
    int idx = k * 65536 + blockIdx.x * 256 + tid;
    merge_one(p[idx], idx, v1, i1, v2, i2);
  }
  sv1[tid] = v1; si1[tid] = i1; sv2[tid] = v2; si2[tid] = i2;
  __syncthreads();
  for (int s = 128; s > 0; s >>= 1) {
    if (tid < s) {
      float a1 = sv1[tid]; int b1 = si1[tid];
      float a2 = sv2[tid]; int b2 = si2[tid];
      merge_one(sv1[tid + s], si1[tid + s], a1, b1, a2, b2);
      merge_one(sv2[tid + s], si2[tid + s], a1, b1, a2, b2);
      sv1[tid] = a1; si1[tid] = b1; sv2[tid] = a2; si2[tid] = b2;
    }
    __syncthreads();
  }
  if (tid == 0) {
    Best2 r; r.v1 = sv1[0]; r.i1 = si1[0]; r.v2 = sv2[0]; r.i2 = si2[0];
    bres[blockIdx.x] = r;
  }
}

__global__ __launch_bounds__(256) void som_reduce2(
    const Best2* __restrict__ bres, const float* __restrict__ v,
    float* __restrict__ out_tail) {
  __shared__ float sv1[256]; __shared__ int si1[256];
  __shared__ float sv2[256]; __shared__ int si2[256];
  __shared__ float svv[256];
  const int tid = threadIdx.x;
  Best2 b = bres[tid];
  float ve = v[tid];
  sv1[tid] = b.v1; si1[tid] = b.i1; sv2[tid] = b.v2; si2[tid] = b.i2;
  svv[tid] = ve * ve;
  __syncthreads();
  for (int s = 128; s > 0; s >>= 1) {
    if (tid < s) {
      float a1 = sv1[tid]; int b1 = si1[tid];
      float a2 = sv2[tid]; int b2 = si2[tid];
      merge_one(sv1[tid + s], si1[tid + s], a1, b1, a2, b2);
      merge_one(sv2[tid + s], si2[tid + s], a1, b1, a2, b2);
      sv1[tid] = a1; si1[tid] = b1; sv2[tid] = a2; si2[tid] = b2;
      svv[tid] += svv[tid + s];
    }
    __syncthreads();
  }
  if (tid == 0) {
    const float vsq = svv[0];
    const int i1 = si1[0], i2 = si2[0];
    const int x = i1 >> 9, y = i1 & 511;
    const int x2 = i2 >> 9, y2 = i2 & 511;
    int dx = x > x2 ? x - x2 : x2 - x; if (dx > XDIM - dx) dx = XDIM - dx;
    int dy = y > y2 ? y - y2 : y2 - y; if (dy > YDIM - dy) dy = YDIM - dy;
    out_tail[0] = (float)x;                              // winner.x
    out_tail[1] = (float)y;                              // winner.y
    out_tail[2] = sqrtf(fmaxf(sv1[0] + vsq, 0.0f));      // qe
    out_tail[3] = sqrtf((float)(dx * dx + dy * dy));     // te
  }
}

// ---------------------------------------------------------------------------
// Phase 3: new_w = w + ALPHA * nbh(cell, winner) * (v - w)
// One float4 per thread, 4 cells per 256-thread block.
// ---------------------------------------------------------------------------
__global__ __launch_bounds__(256) void som_update_kernel(
    const float* __restrict__ w, const float* __restrict__ v,
    const float* __restrict__ winner_f, float* __restrict__ out) {
  const int cell = blockIdx.x * 4 + (threadIdx.x >> 6);
  const int x = cell >> 9, y = cell & 511;
  const int wx = (int)winner_f[0];
  const int wy = (int)winner_f[1];
  int dx = x > wx ? x - wx : wx - x; if (dx > XDIM - dx) dx = XDIM - dx;
  int dy = y > wy ? y - wy : wy - y; if (dy > YDIM - dy) dy = YDIM - dy;
  const float d2 = (float)(dx * dx + dy * dy);
  const float nbh = __expf(-d2 * (1.0f / (2.0f * SIGMA_F * SIGMA_F)));
  const float lr = ALPHA_F * nbh;

  const int dd = (threadIdx.x & 63) * 4;
  const size_t off = (size_t)cell * DDIM + dd;
  float4 a = *(const float4*)(w + off);
  float4 b = *(const float4*)(v + dd);
  float4 o;
  o.x = a.x + lr * (b.x - a.x);
  o.y = a.y + lr * (b.y - a.y);
  o.z = a.z + lr * (b.z - a.z);
  o.w = a.w + lr * (b.w - a.w);
  *(float4*)(out + off) = o;
}

extern "C" void kernel_launch(void* const* d_in, const int* in_sizes, int n_in,
                              void* d_out, int out_size, void* d_ws, size_t ws_size,
                              hipStream_t stream) {
  const float* v = (const float*)d_in[0];   // bu_v  [256]
  const float* w = (const float*)d_in[1];   // w_bu  [512*512*256]
  float* out = (float*)d_out;               // [new_w | x | y | qe | te]

  float* p = out;                              // scratch: dist partials, later overwritten
  float* out_tail = out + (size_t)NCELLS * DDIM;
  Best2* bres = (Best2*)d_ws;                  // 256 * 16 B = 4 KB

  som_dist_kernel<<<NCELLS / ROWS, 256, 0, stream>>>(w, v, p);
  som_reduce1<<<256, 256, 0, stream>>>(p, bres);
  som_reduce2<<<1, 256, 0, stream>>>(bres, v, out_tail);
  som_update_kernel<<<NCELLS / 4, 256, 0, stream>>>(w, v, out_tail, out);
}